// FoldedAnyFLinear_34823594836361
// MI455X (gfx1250) — compile-verified
//
#include <hip/hip_runtime.h>

typedef __attribute__((ext_vector_type(2))) float v2f;
typedef __attribute__((ext_vector_type(8))) float v8f;

#define FOLD   8
#define ISIZE  256
#define OSIZE  256
#define IN_F   2048
#define OUT_F  2048
#define BQ     32        // batch rows per block
#define YSTR   260       // LDS row stride (== 4 mod 64 -> conflict-free b64 A loads)

// ---------------- abs-max reduction ----------------
__global__ void absmax_init(unsigned* ws) {
    if (threadIdx.x == 0 && blockIdx.x == 0) ws[0] = 0u;
}

__global__ __launch_bounds__(256) void absmax_kernel(const float* __restrict__ x,
                                                     unsigned* __restrict__ ws, int n4) {
    const float4* x4 = (const float4*)x;
    unsigned m = 0u;
    for (int idx = blockIdx.x * blockDim.x + threadIdx.x; idx < n4;
         idx += gridDim.x * blockDim.x) {
        float4 v = x4[idx];
        float a = fmaxf(fmaxf(fabsf(v.x), fabsf(v.y)), fmaxf(fabsf(v.z), fabsf(v.w)));
        unsigned u = __float_as_uint(a);   // nonneg float: uint order == float order
        m = (u > m) ? u : m;
    }
    __shared__ unsigned red[256];
    red[threadIdx.x] = m;
    __syncthreads();
    for (int s = 128; s > 0; s >>= 1) {
        if (threadIdx.x < (unsigned)s) {
            unsigned o = red[threadIdx.x + s];
            if (o > red[threadIdx.x]) red[threadIdx.x] = o;
        }
        __syncthreads();
    }
    if (threadIdx.x == 0) atomicMax(ws, red[0]);
}

// ---------------- fused mask + fold-sum + WMMA GEMM ----------------
// block: 256 threads (8 waves). blockIdx.x -> batch chunk of BQ rows, blockIdx.y -> fold f.
// Phase 1: y[r][i] = sum_g masked(x[b0+r, g*256+i] + pe[g,i])  -> LDS
// Phase 2: out[b0.., f*256..] = y @ W_f^T via v_wmma_f32_16x16x4_f32
__global__ __launch_bounds__(256, 2) void folded_kernel(
        const float* __restrict__ x, const float* __restrict__ W,
        const float* __restrict__ pe, const float* __restrict__ mnr,
        const float* __restrict__ mxr, const unsigned* __restrict__ ws,
        float* __restrict__ out) {
    __shared__ float y_s[BQ * YSTR];

    const int f   = blockIdx.y;
    const int b0  = blockIdx.x * BQ;
    const int tid = threadIdx.x;
    const float s = __uint_as_float(ws[0]);   // == max|x|  (s*QMAX in reference)

    // ---- phase 1: each thread owns column i = tid ----
    const int i = tid;
    const float lo = mnr[f * ISIZE + i] * s;
    const float hi = mxr[f * ISIZE + i] * s;
    float peg[FOLD];
#pragma unroll
    for (int g = 0; g < FOLD; ++g) peg[g] = pe[g * ISIZE + i];

    for (int r = 0; r < BQ; ++r) {
        const float* xrow = x + (size_t)(b0 + r) * IN_F;
        float acc = 0.f;
#pragma unroll
        for (int g = 0; g < FOLD; ++g) {
            float v = xrow[g * ISIZE + i] + peg[g];
            acc += (v >= lo && v <= hi) ? v : 0.f;   // mask strict: drop v<lo or v>hi
        }
        y_s[r * YSTR + i] = acc;
    }
    __syncthreads();

    // ---- phase 2: WMMA GEMM, wave w covers N slab [w*32, w*32+32) ----
    const int wave  = tid >> 5;
    const int lane  = tid & 31;
    const int lrow  = lane & 15;   // M (for A) / N (for B,D)
    const int lhalf = lane >> 4;   // K half-select for A/B, M+8 for D
    const int nbase = wave * 32;
    const float* Wf = W + (size_t)f * OSIZE * ISIZE;

    v8f acc[2][2] = {};            // 2 M-tiles x 2 N-tiles of 16x16 f32

#pragma unroll 4
    for (int k = 0; k < ISIZE; k += 4) {
        const int koff = k + (lhalf << 1);   // ISA: VGPR = K within half, lanes16-31 = +K/2
        v2f a[2], b[2];
#pragma unroll
        for (int mt = 0; mt < 2; ++mt)
            a[mt] = *(const v2f*)&y_s[(mt * 16 + lrow) * YSTR + koff];
#pragma unroll
        for (int nt = 0; nt < 2; ++nt)
            b[nt] = *(const v2f*)(Wf + (size_t)(nbase + nt * 16 + lrow) * ISIZE + koff);
#pragma unroll
        for (int mt = 0; mt < 2; ++mt)
#pragma unroll
            for (int nt = 0; nt < 2; ++nt)
                acc[mt][nt] = __builtin_amdgcn_wmma_f32_16x16x4_f32(
                    false, a[mt], false, b[nt], (short)0, acc[mt][nt], false, false);
    }

    // ---- store D: VGPR r -> row M = r + 8*lhalf, lane -> N = lrow ----
#pragma unroll
    for (int mt = 0; mt < 2; ++mt)
#pragma unroll
        for (int nt = 0; nt < 2; ++nt)
#pragma unroll
            for (int r = 0; r < 8; ++r) {
                const int row = b0 + mt * 16 + r + lhalf * 8;
                const int col = f * OSIZE + nbase + nt * 16 + lrow;
                out[(size_t)row * OUT_F + col] = acc[mt][nt][r];
            }
}

extern "C" void kernel_launch(void* const* d_in, const int* in_sizes, int n_in,
                              void* d_out, int out_size, void* d_ws, size_t ws_size,
                              hipStream_t stream) {
    const float* x   = (const float*)d_in[0];   // [B, 2048]
    const float* W   = (const float*)d_in[1];   // [8, 256, 256]
    const float* pe  = (const float*)d_in[2];   // [1, 8, 256]
    const float* mnr = (const float*)d_in[3];   // [1, 8, 256]
    const float* mxr = (const float*)d_in[4];   // [1, 8, 256]
    float* out  = (float*)d_out;
    unsigned* ws = (unsigned*)d_ws;

    const int B  = in_sizes[0] / IN_F;          // 8192
    const int n4 = (B * IN_F) / 4;

    absmax_init<<<1, 1, 0, stream>>>(ws);
    absmax_kernel<<<1024, 256, 0, stream>>>(x, ws, n4);

    dim3 grid(B / BQ, FOLD);
    folded_kernel<<<grid, 256, 0, stream>>>(x, W, pe, mnr, mxr, ws, out);
}